// QuantumConvLayer_57466662420514
// MI455X (gfx1250) — compile-verified
//
#include <hip/hip_runtime.h>
#include <math.h>

typedef float v2f __attribute__((ext_vector_type(2)));
typedef float v8f __attribute__((ext_vector_type(8)));

#define BATCH 8
#define HNUM 96
#define WNUM 96
#define CCH 16
#define OH 95
#define OW 95
#define NPATCH (BATCH * OH * OW)   // 72200
#define NFILT 8

// ---------------------------------------------------------------------------
// Kernel 1: build the 8 constant 16x16 real quadratic-form matrices
//   M_f = Re(L_f)^T diag(Z) Re(L_f) + Im(L_f)^T diag(Z) Im(L_f)
// where L_f = Circuit_f * CNOTchain.  One thread = (filter, basis column).
// ---------------------------------------------------------------------------
__global__ __launch_bounds__(128) void build_M(const float* __restrict__ w,
                                               float* __restrict__ M)
{
    __shared__ float Sre[NFILT][16][16];   // Sre[f][col][k] = Re L_f[k][col]
    __shared__ float Sim[NFILT][16][16];

    const int t   = threadIdx.x;           // 0..127
    const int f   = t >> 4;
    const int col = t & 15;

    float re[16], im[16];
#pragma unroll
    for (int k = 0; k < 16; ++k) { re[k] = (k == col) ? 1.f : 0.f; im[k] = 0.f; }

    // CNOT chain: control q -> target q+1 ; qubit q lives in bit (3-q)
#pragma unroll
    for (int q = 0; q < 3; ++q) {
        const int cm = 1 << (3 - q), tm = 1 << (2 - q);
        float nr[16], ni[16];
#pragma unroll
        for (int k = 0; k < 16; ++k) {
            const int s = (k & cm) ? (k ^ tm) : k;
            nr[k] = re[s]; ni[k] = im[s];
        }
#pragma unroll
        for (int k = 0; k < 16; ++k) { re[k] = nr[k]; im[k] = ni[k]; }
    }

    const float* wf = w + f * 24;
#pragma unroll
    for (int layer = 0; layer < 2; ++layer) {
        const int base = layer * 12;
#pragma unroll
        for (int q = 0; q < 4; ++q) {
            const int m = 1 << (3 - q);
            float th, c, s;
            // RX(theta): [[c, -i s],[-i s, c]]
            th = 0.5f * wf[base + q * 3 + 0]; c = cosf(th); s = sinf(th);
#pragma unroll
            for (int k = 0; k < 16; ++k) if (!(k & m)) {
                const int k1 = k | m;
                const float r0 = re[k], i0 = im[k], r1 = re[k1], i1 = im[k1];
                re[k]  = c * r0 + s * i1;  im[k]  = c * i0 - s * r1;
                re[k1] = s * i0 + c * r1;  im[k1] = c * i1 - s * r0;
            }
            // RY(theta): [[c, -s],[s, c]]
            th = 0.5f * wf[base + q * 3 + 1]; c = cosf(th); s = sinf(th);
#pragma unroll
            for (int k = 0; k < 16; ++k) if (!(k & m)) {
                const int k1 = k | m;
                const float r0 = re[k], i0 = im[k], r1 = re[k1], i1 = im[k1];
                re[k]  = c * r0 - s * r1;  im[k]  = c * i0 - s * i1;
                re[k1] = s * r0 + c * r1;  im[k1] = s * i0 + c * i1;
            }
            // RZ(theta): diag(e^{-i th/2}, e^{+i th/2})
            th = 0.5f * wf[base + q * 3 + 2]; c = cosf(th); s = sinf(th);
#pragma unroll
            for (int k = 0; k < 16; ++k) {
                const float r0 = re[k], i0 = im[k];
                if (!(k & m)) { re[k] = c * r0 + s * i0; im[k] = c * i0 - s * r0; }
                else          { re[k] = c * r0 - s * i0; im[k] = c * i0 + s * r0; }
            }
        }
        // CRY entanglers on (0,1) and (2,3), theta = w[base+q]
#pragma unroll
        for (int q = 0; q < 4; q += 2) {
            const float th2 = 0.5f * wf[base + q];
            const float c = cosf(th2), s = sinf(th2);
            const int cm = 1 << (3 - q), tm = 1 << (2 - q);
#pragma unroll
            for (int k = 0; k < 16; ++k) if ((k & cm) && !(k & tm)) {
                const int k1 = k | tm;
                const float r0 = re[k], i0 = im[k], r1 = re[k1], i1 = im[k1];
                re[k]  = c * r0 - s * r1;  im[k]  = c * i0 - s * i1;
                re[k1] = s * r0 + c * r1;  im[k1] = s * i0 + c * i1;
            }
        }
    }

#pragma unroll
    for (int k = 0; k < 16; ++k) { Sre[f][col][k] = re[k]; Sim[f][col][k] = im[k]; }
    __syncthreads();

    // M_f[i][col] = sum_k Z[k] * (ReL[k][i] ReL[k][col] + ImL[k][i] ImL[k][col])
    for (int i = 0; i < 16; ++i) {
        float acc = 0.f;
#pragma unroll
        for (int k = 0; k < 16; ++k) {
            const float z = 1.f - 0.5f * (float)__popc(k);  // Z_MEAN[k]
            acc += z * (Sre[f][i][k] * Sre[f][col][k] +
                        Sim[f][i][k] * Sim[f][col][k]);
        }
        M[f * 256 + i * 16 + col] = acc;   // row-major: M[f][row][colk]
    }
}

// ---------------------------------------------------------------------------
// Kernel 2: per-patch psi prep + out[p,f] = psi^T M_f psi via fp32 WMMA.
// 256 threads = 8 waves; each wave handles a tile of 16 patches.
// Orientation: Y' = M_f (A, from LDS) x Psi^T (B, register-resident) so that
// the C layout lands state-major per patch-lane -> final dot is lane-local.
// Filter loop fully unrolled: outf[] stays in statically-indexed VGPRs
// (no v_movreld / m0), giving a straight line of 32 pipelined WMMAs.
// ---------------------------------------------------------------------------
__global__ __launch_bounds__(256) void qconv_main(const float* __restrict__ x,
                                                  const float* __restrict__ Mg,
                                                  float* __restrict__ out)
{
    __shared__ float Mlds[NFILT * 256];      // 8 KB: all filter matrices

    const int tid = threadIdx.x;
    for (int i = tid; i < NFILT * 256; i += 256) Mlds[i] = Mg[i];
    __syncthreads();

    const int wave = tid >> 5;
    const int lane = tid & 31;
    const int h    = lane >> 4;      // half-wave id
    const int n    = lane & 15;      // patch-within-tile / WMMA N index

    const int pbase = (blockIdx.x * 8 + wave) * 16;
    const int p     = pbase + n;
    const int pc    = (p < NPATCH) ? p : (NPATCH - 1);     // clamp (stores guarded)
    const int b     = pc / (OH * OW);
    const int rem   = pc - b * (OH * OW);
    const int r     = rem / OW;
    const int cc    = rem - r * OW;

    // Lanes L and L+16 cooperatively load patch (pbase+n): half h loads row fr=h,
    // i.e. 2x16 contiguous channels = 8 float4s (fully coalesced b128 loads).
    const float4* bp = reinterpret_cast<const float4*>(
        x + ((size_t)(b * HNUM + (r + h)) * WNUM + cc) * CCH);
    float  ss    = 0.f;
    float4 first = bp[0];                    // elements (fr=0,fc=0,ch0..3) on h==0
#pragma unroll
    for (int i = 0; i < 8; ++i) {
        const float4 v = bp[i];
        ss += v.x * v.x + v.y * v.y + v.z * v.z + v.w * v.w;
    }
    ss += __shfl_xor(ss, 16);                // full 64-element sum of squares
    const float a0 = __shfl(first.x, n);     // broadcast h==0 lane's first 4 elems
    const float a1 = __shfl(first.y, n);
    const float a2 = __shfl(first.z, n);
    const float a3 = __shfl(first.w, n);

    const float rinv = rsqrtf(fmaxf(ss, 1e-12f));
    const float t0 = 0.5f * a0 * rinv, t1 = 0.5f * a1 * rinv;
    const float t2 = 0.5f * a2 * rinv, t3 = 0.5f * a3 * rinv;
    const float c0 = cosf(t0), s0 = sinf(t0);
    const float c1 = cosf(t1), s1 = sinf(t1);
    const float c2 = cosf(t2), s2 = sinf(t2);
    const float c3 = cosf(t3), s3 = sinf(t3);
    // psi[k] = prod_q (bit_q(k) ? sin : cos); qubit q = bit (3-q) of k
    const float g01[4] = { c0 * c1, c0 * s1, s0 * c1, s0 * s1 };
    const float g23[4] = { c2 * c3, c2 * s3, s2 * c3, s2 * s3 };
    float psi[16];
#pragma unroll
    for (int k = 0; k < 16; ++k) psi[k] = g01[k >> 2] * g23[k & 3];

    // B-layout of Psi^T (4 chunks of 4x16): lane (h,n), reg v: K = 4*ck + 2*h + v,
    // element = psi_{patch n}[K].  Runtime h handled by per-lane select
    // (indices stay compile-time so psi[] remains in registers).
    float breg[8];
#pragma unroll
    for (int ck = 0; ck < 4; ++ck) {
        breg[2 * ck + 0] = h ? psi[4 * ck + 2] : psi[4 * ck + 0];
        breg[2 * ck + 1] = h ? psi[4 * ck + 3] : psi[4 * ck + 1];
    }
    // Psi elements matching the C layout rows (M = v + 8h) for the final dot.
    float psid[8];
#pragma unroll
    for (int v = 0; v < 8; ++v) psid[v] = h ? psi[v + 8] : psi[v];

    float outf[NFILT];
#pragma unroll
    for (int f = 0; f < NFILT; ++f) {
        v8f acc = { 0.f, 0.f, 0.f, 0.f, 0.f, 0.f, 0.f, 0.f };
#pragma unroll
        for (int ck = 0; ck < 4; ++ck) {
            // A-layout of M_f (16x4 chunk): lane row m = n, K = 4*ck + 2*h + v.
            // Two consecutive K per lane -> single 64-bit LDS load.
            const float2 av = *reinterpret_cast<const float2*>(
                &Mlds[f * 256 + n * 16 + 4 * ck + 2 * h]);
            const v2f A = { av.x, av.y };
            const v2f B = { breg[2 * ck + 0], breg[2 * ck + 1] };
            acc = __builtin_amdgcn_wmma_f32_16x16x4_f32(
                      false, A, false, B, (short)0, acc, false, false);
        }
        // acc[v] = Y'[state v+8h][patch n]; dot with psi over this half's states.
        float dot = 0.f;
#pragma unroll
        for (int v = 0; v < 8; ++v) dot = fmaf(psid[v], acc[v], dot);
        outf[f] = dot;
    }

    // Combine the two half-wave state partitions: one xor-16 shuffle per filter.
#pragma unroll
    for (int f = 0; f < NFILT; ++f) outf[f] += __shfl_xor(outf[f], 16);

    if (h == 0 && p < NPATCH) {
        float4* op = reinterpret_cast<float4*>(out + (size_t)p * NFILT);
        op[0] = make_float4(outf[0], outf[1], outf[2], outf[3]);
        op[1] = make_float4(outf[4], outf[5], outf[6], outf[7]);
    }
}

extern "C" void kernel_launch(void* const* d_in, const int* in_sizes, int n_in,
                              void* d_out, int out_size, void* d_ws, size_t ws_size,
                              hipStream_t stream)
{
    (void)in_sizes; (void)n_in; (void)out_size; (void)ws_size;
    const float* x = (const float*)d_in[0];        // (8,96,96,16) f32
    const float* w = (const float*)d_in[1];        // (8,24) f32
    float* out = (float*)d_out;                    // (8,95,95,8) f32
    float* M   = (float*)d_ws;                     // 8 filters * 256 f32 = 8 KB

    build_M<<<1, 128, 0, stream>>>(w, M);
    const int nblocks = (NPATCH + 127) / 128;      // 8 waves * 16 patches per block
    qconv_main<<<nblocks, 256, 0, stream>>>(x, M, out);
}